// MinkUNet_35442070126791
// MI455X (gfx1250) — compile-verified
//
#include <hip/hip_runtime.h>
#include <hip/hip_bf16.h>

typedef __attribute__((ext_vector_type(2))) float v2f;
typedef __attribute__((ext_vector_type(4))) float f4;
typedef __attribute__((ext_vector_type(8))) float v8f;

#define CCH 64            // channels
#define LSTR 68           // padded LDS row stride (floats): conflict-free, 16B aligned
#define TILES 4           // 16-row tiles per block
#define RPB   (TILES * 16)

// ---------------------------------------------------------------------------
// Sparse conv: out[64 rows x 64 cols] per block via V_WMMA_F32_16X16X4_F32.
// Block = 128 threads = 4 waves; wave w computes output cols [16w,16w+16) for
// all 4 row-tiles (B fragments reused 4x, 4 independent WMMA chains).
// Gathers for k+1 are staged in registers while WMMAs for k execute.
// ---------------------------------------------------------------------------
__global__ __launch_bounds__(128)
void conv_wmma_kernel(const float* __restrict__ x, const int* __restrict__ nbr,
                      const float* __restrict__ W, float* __restrict__ out,
                      int n, int K)
{
    __shared__ float lds[RPB * LSTR];

    const int t    = threadIdx.x;        // 0..127
    const int wave = t >> 5;             // 0..3
    const int lane = t & 31;
    const int row0 = blockIdx.x * RPB;

    const int gm  = t >> 3;              // base gather row 0..15
    const int gch = (t & 7) * 8;         // gather channel start

    const int Ml = lane & 15;
    const int kb = (lane < 16) ? 0 : 2;
    const int co = wave * 16 + Ml;       // this lane's output column (B/C layout)

    v8f acc0 = {}, acc1 = {}, acc2 = {}, acc3 = {};
    f4 sv[TILES][2];                     // staged gather registers

    auto stage = [&](int k) {
        #pragma unroll
        for (int j = 0; j < TILES; ++j) {
            const int row = row0 + gm + j * 16;
            int gi = (row < n) ? nbr[(size_t)row * K + k] : -1;
            f4 v0 = {}, v1 = {};
            if (gi >= 0) {
                const float* src = x + (size_t)gi * CCH + gch;
                v0 = *(const f4*)(src);
                v1 = *(const f4*)(src + 4);
            }
            sv[j][0] = v0;
            sv[j][1] = v1;
        }
    };

    stage(0);
    for (int k = 0; k < K; ++k) {
        __syncthreads();                 // previous iteration's LDS fully consumed
        #pragma unroll
        for (int j = 0; j < TILES; ++j) {
            *(f4*)&lds[(gm + j * 16) * LSTR + gch]     = sv[j][0];
            *(f4*)&lds[(gm + j * 16) * LSTR + gch + 4] = sv[j][1];
        }
        __syncthreads();

        const float* Wkp = W + (size_t)k * CCH * CCH;
        if (k + 1 < K) {
            stage(k + 1);                // gather latency overlaps WMMAs below
            __builtin_prefetch(W + (size_t)(k + 1) * CCH * CCH + t * 32, 0, 1);
        }

        #pragma unroll
        for (int kk = 0; kk < 16; ++kk) {
            const int ci = kk * 4 + kb;
            v2f b, a;
            b[0] = Wkp[ci * CCH + co];           // B[ci][co], B[ci+1][co]
            b[1] = Wkp[(ci + 1) * CCH + co];

            a[0] = lds[(0 * 16 + Ml) * LSTR + ci];
            a[1] = lds[(0 * 16 + Ml) * LSTR + ci + 1];
            acc0 = __builtin_amdgcn_wmma_f32_16x16x4_f32(false, a, false, b, (short)0, acc0, false, false);
            a[0] = lds[(1 * 16 + Ml) * LSTR + ci];
            a[1] = lds[(1 * 16 + Ml) * LSTR + ci + 1];
            acc1 = __builtin_amdgcn_wmma_f32_16x16x4_f32(false, a, false, b, (short)0, acc1, false, false);
            a[0] = lds[(2 * 16 + Ml) * LSTR + ci];
            a[1] = lds[(2 * 16 + Ml) * LSTR + ci + 1];
            acc2 = __builtin_amdgcn_wmma_f32_16x16x4_f32(false, a, false, b, (short)0, acc2, false, false);
            a[0] = lds[(3 * 16 + Ml) * LSTR + ci];
            a[1] = lds[(3 * 16 + Ml) * LSTR + ci + 1];
            acc3 = __builtin_amdgcn_wmma_f32_16x16x4_f32(false, a, false, b, (short)0, acc3, false, false);
        }
    }

    // ---- write back: VGPR r -> row (r + 0/8) of each tile, col = co --------
    const int rb = (lane < 16) ? 0 : 8;
    #pragma unroll
    for (int r = 0; r < 8; ++r) {
        int row = row0 + 0 * 16 + rb + r;
        if (row < n) out[(size_t)row * CCH + co] = acc0[r];
    }
    #pragma unroll
    for (int r = 0; r < 8; ++r) {
        int row = row0 + 1 * 16 + rb + r;
        if (row < n) out[(size_t)row * CCH + co] = acc1[r];
    }
    #pragma unroll
    for (int r = 0; r < 8; ++r) {
        int row = row0 + 2 * 16 + rb + r;
        if (row < n) out[(size_t)row * CCH + co] = acc2[r];
    }
    #pragma unroll
    for (int r = 0; r < 8; ++r) {
        int row = row0 + 3 * 16 + rb + r;
        if (row < n) out[(size_t)row * CCH + co] = acc3[r];
    }
}

// ---------------------------------------------------------------------------
// BN statistics: deterministic two-pass per-channel sum / sumsq.
// ---------------------------------------------------------------------------
__global__ void stats_partial_kernel(const float* __restrict__ v,
                                     float* __restrict__ partial, int n)
{
    const int c  = threadIdx.x & 63;
    const int rg = threadIdx.x >> 6;     // 0..3
    float s = 0.f, q = 0.f;
    for (int r = blockIdx.x * 4 + rg; r < n; r += gridDim.x * 4) {
        float val = v[(size_t)r * CCH + c];
        s += val; q += val * val;
    }
    __shared__ float ls[256], lq[256];
    ls[threadIdx.x] = s; lq[threadIdx.x] = q;
    __syncthreads();
    if (threadIdx.x < 64) {
        s = ls[threadIdx.x] + ls[threadIdx.x + 64] + ls[threadIdx.x + 128] + ls[threadIdx.x + 192];
        q = lq[threadIdx.x] + lq[threadIdx.x + 64] + lq[threadIdx.x + 128] + lq[threadIdx.x + 192];
        partial[(size_t)blockIdx.x * 128 + threadIdx.x]      = s;
        partial[(size_t)blockIdx.x * 128 + 64 + threadIdx.x] = q;
    }
}

__global__ void stats_reduce_kernel(const float* __restrict__ partial,
                                    float* __restrict__ sums, float* __restrict__ sumsq,
                                    int nblk)
{
    const int t = threadIdx.x;           // 0..127
    float a = 0.f;
    for (int b = 0; b < nblk; ++b) a += partial[(size_t)b * 128 + t];
    if (t < 64) sums[t] = a; else sumsq[t - 64] = a;
}

__global__ void bn_apply_kernel(const float* __restrict__ in, float* __restrict__ dst,
                                const float* __restrict__ sums, const float* __restrict__ sumsq,
                                const float* __restrict__ g, const float* __restrict__ b,
                                int n, int do_relu, float invN)
{
    size_t i = (size_t)blockIdx.x * blockDim.x + threadIdx.x;
    if (i >= (size_t)n * CCH) return;
    int c = (int)(i & 63);
    float m   = sums[c] * invN;
    float var = sumsq[c] * invN - m * m;
    float y = (in[i] - m) * rsqrtf(var + 1e-5f) * g[c] + b[c];
    if (do_relu) y = fmaxf(y, 0.f);
    dst[i] = y;
}

// ---------------------------------------------------------------------------
// Channel attention: per-batch segment sum / max over xcat = [kx0|kx1|kx2].
// Order-preserving float->uint encoding so atomicMax(u32) works for any sign.
// ---------------------------------------------------------------------------
__device__ __forceinline__ unsigned encf(float f) {
    unsigned u = __float_as_uint(f);
    return (u & 0x80000000u) ? ~u : (u | 0x80000000u);
}
__device__ __forceinline__ float decf(unsigned k) {
    unsigned u = (k & 0x80000000u) ? (k ^ 0x80000000u) : ~k;
    return __uint_as_float(u);
}
#define ENC_NEG_INF 0x007FFFFFu          // encf(-inf)

__global__ void fill_u32_kernel(unsigned* __restrict__ p, unsigned v, int n) {
    int i = blockIdx.x * blockDim.x + threadIdx.x;
    if (i < n) p[i] = v;
}

__global__ void seg_partial_kernel(const float* __restrict__ kx0, const float* __restrict__ kx1,
                                   const float* __restrict__ kx2, const int* __restrict__ bidx,
                                   float* __restrict__ psum, float* __restrict__ pcnt,
                                   unsigned* __restrict__ mx_enc, int n)
{
    const int blk = blockIdx.x;
    const int r0  = blk * 128;
    const int r1  = (r0 + 128 < n) ? r0 + 128 : n;
    const int c   = threadIdx.x;

    if (c >= 192) {
        if (c == 192) {
            int c0 = 0;
            for (int r = r0; r < r1; ++r) c0 += (bidx[r] == 0);
            pcnt[blk * 2 + 0] = (float)c0;
            pcnt[blk * 2 + 1] = (float)((r1 - r0) - c0);
        }
        return;
    }
    const float* src = (c < 64) ? kx0 : (c < 128) ? kx1 : kx2;
    const int cc = c & 63;
    float s0 = 0.f, s1 = 0.f, m0 = -3.4e38f, m1 = -3.4e38f;
    bool a0 = false, a1 = false;
    for (int r = r0; r < r1; ++r) {
        float v = src[(size_t)r * CCH + cc];
        if (bidx[r] == 0) { s0 += v; m0 = fmaxf(m0, v); a0 = true; }
        else              { s1 += v; m1 = fmaxf(m1, v); a1 = true; }
    }
    psum[(size_t)blk * 384 + c]       = s0;
    psum[(size_t)blk * 384 + 192 + c] = s1;
    if (a0) atomicMax(&mx_enc[c],       encf(m0));
    if (a1) atomicMax(&mx_enc[192 + c], encf(m1));
}

__global__ void seg_reduce_kernel(const float* __restrict__ psum, const float* __restrict__ pcnt,
                                  float* __restrict__ avg_sum, float* __restrict__ cnt, int nblk)
{
    const int t = threadIdx.x;           // 0..383
    float a = 0.f;
    for (int b = 0; b < nblk; ++b) a += psum[(size_t)b * 384 + t];
    avg_sum[t] = a;
    if (t < 2) {
        float cc = 0.f;
        for (int b = 0; b < nblk; ++b) cc += pcnt[b * 2 + t];
        cnt[t] = cc;
    }
}

// Tiny FC: att = sigmoid(fc(avg) + fc(max)), fc: 192 -> 12 (relu) -> 192.
__global__ void fc_kernel(const float* __restrict__ avg_sum, const unsigned* __restrict__ mx_enc,
                          const float* __restrict__ cnt,
                          const float* __restrict__ fc1_w, const float* __restrict__ fc1_b,
                          const float* __restrict__ fc2_w, const float* __restrict__ fc2_b,
                          float* __restrict__ att)
{
    __shared__ float sAvg[2][192], sMx[2][192], sHa[2][12], sHm[2][12];
    const int t = threadIdx.x;
    if (t < 192) {
        for (int b = 0; b < 2; ++b) {
            sAvg[b][t] = avg_sum[b * 192 + t] / cnt[b];
            sMx[b][t]  = decf(mx_enc[b * 192 + t]);
        }
    }
    __syncthreads();
    if (t < 24) {
        const int b = t / 12, r = t % 12;
        float sa = fc1_b[r], sm = fc1_b[r];
        for (int j = 0; j < 192; ++j) {
            float w = fc1_w[j * 12 + r];
            sa += sAvg[b][j] * w;
            sm += sMx[b][j]  * w;
        }
        sHa[b][r] = fmaxf(sa, 0.f);
        sHm[b][r] = fmaxf(sm, 0.f);
    }
    __syncthreads();
    if (t < 192) {
        for (int b = 0; b < 2; ++b) {
            float za = fc2_b[t], zm = fc2_b[t];
            for (int r = 0; r < 12; ++r) {
                float w = fc2_w[r * 192 + t];
                za += sHa[b][r] * w;
                zm += sHm[b][r] * w;
            }
            float s = za + zm;
            att[b * 192 + t] = 1.f / (1.f + expf(-s));
        }
    }
}

// out[n,c] = relu(x + sum_i w_i * (kx_i[n,c] + att[b,3c+i] * xcat[n,3c+i]))
__global__ void final_kernel(const float* __restrict__ x,
                             const float* __restrict__ kx0, const float* __restrict__ kx1,
                             const float* __restrict__ kx2, const int* __restrict__ bidx,
                             const float* __restrict__ att, const float* __restrict__ ksw,
                             float* __restrict__ out, int n)
{
    size_t i = (size_t)blockIdx.x * blockDim.x + threadIdx.x;
    if (i >= (size_t)n * CCH) return;
    const int row = (int)(i >> 6), c = (int)(i & 63);
    const int b = bidx[row];
    float res = x[i];
    #pragma unroll
    for (int t = 0; t < 3; ++t) {
        float w = 1.f / (1.f + expf(-ksw[t]));
        int j = c * 3 + t;                      // xcat column after reshape(-1,C,NX)
        int sj = j >> 6, jc = j & 63;
        const float* kp = (sj == 0) ? kx0 : (sj == 1) ? kx1 : kx2;
        float xc = kp[(size_t)row * CCH + jc];
        const float* kq = (t == 0) ? kx0 : (t == 1) ? kx1 : kx2;
        float ki = kq[(size_t)row * CCH + c];
        res += w * (ki + xc * att[b * 192 + j]);
    }
    out[i] = fmaxf(res, 0.f);
}

// ---------------------------------------------------------------------------
extern "C" void kernel_launch(void* const* d_in, const int* in_sizes, int n_in,
                              void* d_out, int out_size, void* d_ws, size_t ws_size,
                              hipStream_t stream)
{
    const float* feats = (const float*)d_in[0];
    const int*   bidx  = (const int*)  d_in[1];
    const int*   nbr   = (const int*)  d_in[2];
    const float* W1    = (const float*)d_in[3];
    const float* g1    = (const float*)d_in[4];
    const float* b1    = (const float*)d_in[5];
    const float* W2    = (const float*)d_in[6];
    const float* g2    = (const float*)d_in[7];
    const float* b2    = (const float*)d_in[8];
    const float* Wk    = (const float*)d_in[9];
    const float* gk    = (const float*)d_in[10];
    const float* bk    = (const float*)d_in[11];
    const float* fc1w  = (const float*)d_in[12];
    const float* fc1b  = (const float*)d_in[13];
    const float* fc2w  = (const float*)d_in[14];
    const float* fc2b  = (const float*)d_in[15];
    const float* ksw   = (const float*)d_in[16];

    const int n = in_sizes[0] / CCH;            // 80000
    const int K = in_sizes[2] / n;              // 27
    const size_t NC = (size_t)n * CCH;

    float* ws   = (float*)d_ws;
    float* tmp  = ws;                 // raw conv output
    float* bufA = ws + NC;            // h1 then h2
    float* kx0  = ws + 2 * NC;
    float* kx1  = ws + 3 * NC;
    float* kx2  = ws + 4 * NC;
    size_t off  = 5 * NC;
    float*    sums    = ws + off;                 off += 64;
    float*    sumsq   = ws + off;                 off += 64;
    float*    avg_sum = ws + off;                 off += 384;
    float*    cntp    = ws + off;                 off += 2;
    float*    att     = ws + off;                 off += 384;
    unsigned* mx_enc  = (unsigned*)(ws + off);    off += 384;
    float*    spart   = ws + off;                 off += 512 * 128;
    const int nseg    = (n + 127) / 128;
    float*    gpart   = ws + off;                 off += (size_t)nseg * 384;
    float*    gcnt    = ws + off;                 off += (size_t)nseg * 2;

    const int   cblocks = (n + RPB - 1) / RPB;
    const int   eblocks = (int)((NC + 255) / 256);
    const float invN    = 1.0f / (float)n;

    auto conv_bn = [&](const float* src, const float* W, const float* g,
                       const float* b, float* dst, int relu) {
        conv_wmma_kernel<<<cblocks, 128, 0, stream>>>(src, nbr, W, tmp, n, K);
        stats_partial_kernel<<<512, 256, 0, stream>>>(tmp, spart, n);
        stats_reduce_kernel<<<1, 128, 0, stream>>>(spart, sums, sumsq, 512);
        bn_apply_kernel<<<eblocks, 256, 0, stream>>>(tmp, dst, sums, sumsq, g, b, n, relu, invN);
    };

    const size_t KCC = (size_t)K * CCH * CCH;
    conv_bn(feats, W1,           g1,       b1,       bufA, 1);   // h1
    conv_bn(bufA,  W2,           g2,       b2,       bufA, 0);   // h2
    conv_bn(bufA,  Wk + 0 * KCC, gk,       bk,       kx0,  1);
    conv_bn(kx0,   Wk + 1 * KCC, gk + 64,  bk + 64,  kx1,  1);
    conv_bn(kx1,   Wk + 2 * KCC, gk + 128, bk + 128, kx2,  1);

    fill_u32_kernel<<<2, 256, 0, stream>>>(mx_enc, ENC_NEG_INF, 384);
    seg_partial_kernel<<<nseg, 256, 0, stream>>>(kx0, kx1, kx2, bidx, gpart, gcnt, mx_enc, n);
    seg_reduce_kernel<<<1, 384, 0, stream>>>(gpart, gcnt, avg_sum, cntp, nseg);
    fc_kernel<<<1, 256, 0, stream>>>(avg_sum, mx_enc, cntp, fc1w, fc1b, fc2w, fc2b, att);
    final_kernel<<<eblocks, 256, 0, stream>>>(feats, kx0, kx1, kx2, bidx, att, ksw,
                                              (float*)d_out, n);
}